// SparseConvCausalAttention_11209864643041
// MI455X (gfx1250) — compile-verified
//
#include <hip/hip_runtime.h>
#include <math.h>

// ---------------- problem constants ----------------
#define BATCH     4
#define SEQ       1280
#define DIM       512
#define HEADS     8
#define DHEAD     64
#define INNER     512          // HEADS*DHEAD
#define NQKV      1536         // 3*INNER
#define TEXTLEN   256
#define IMGLEN    1024
#define IMGSZ     32
#define KK        25           // 5x5
#define PADR      2
#define BH        32           // BATCH*HEADS
#define QSCALE    0.125f       // 64^-0.5
#define NEGMAX    (-3.402823466e38f)

typedef __attribute__((ext_vector_type(2))) float v2f;
typedef __attribute__((ext_vector_type(8))) float v8f;

// ---------------- WMMA f32 16x16x4 helpers ----------------
// D = A(16x4) * B(4x16) + C(16x16), fp32 throughout (matches fp32 reference).
__device__ __forceinline__ v8f wmma4(v2f a, v2f b, v8f c) {
  return __builtin_amdgcn_wmma_f32_16x16x4_f32(
      /*neg_a=*/false, a, /*neg_b=*/false, b,
      /*c_mod=*/(short)0, c, /*reuse_a=*/false, /*reuse_b=*/false);
}

// A fragment: 16x4 (MxK).  lanes 0-15: M=lane, VGPR0=K0,VGPR1=K1.
//                          lanes16-31: M=lane-16, VGPR0=K2,VGPR1=K3.
__device__ __forceinline__ v2f ldA(const float* __restrict__ A, int lda,
                                   int row0, int k0, int lane) {
  const int g = lane >> 4;
  const float* p = A + (size_t)(row0 + (lane & 15)) * lda + (k0 + 2 * g);
  v2f r; r.x = p[0]; r.y = p[1];
  return r;
}
// B fragment: 4x16 (KxN) row-major B. VGPR0: K0(l0-15)/K2(l16-31); VGPR1: K1/K3.
__device__ __forceinline__ v2f ldB(const float* __restrict__ B, int ldb,
                                   int k0, int col0, int lane) {
  const int g = lane >> 4;
  const float* p = B + (size_t)(k0 + 2 * g) * ldb + (col0 + (lane & 15));
  v2f r; r.x = p[0]; r.y = p[ldb];
  return r;
}
// B fragment when B is stored transposed (Bt[n][k] => B[k][n]): contiguous pair.
__device__ __forceinline__ v2f ldBT(const float* __restrict__ Bt, int ldt,
                                    int k0, int col0, int lane) {
  const int g = lane >> 4;
  const float* p = Bt + (size_t)(col0 + (lane & 15)) * ldt + (k0 + 2 * g);
  v2f r; r.x = p[0]; r.y = p[1];
  return r;
}

// =====================================================================
// Kernel 1: QKV projection.  x[5120,512] @ Wqkv[512,1536] -> Q,K,V [bh][n][64]
// grid (40, 24), block 256 (8 waves); wave tile = 16 rows x 64 cols.
// =====================================================================
__global__ __launch_bounds__(256)
void k_qkv(const float* __restrict__ x, const float* __restrict__ Wqkv,
           float* __restrict__ Q, float* __restrict__ K, float* __restrict__ V) {
  const int lane = threadIdx.x & 31;
  const int wave = threadIdx.x >> 5;
  const int row0 = blockIdx.x * 128 + wave * 16;   // 0..5104
  const int col0 = blockIdx.y * 64;                // 0..1472

  v8f acc[4] = {v8f{}, v8f{}, v8f{}, v8f{}};
  for (int k0 = 0; k0 < DIM; k0 += 4) {
    v2f a = ldA(x, DIM, row0, k0, lane);
#pragma unroll
    for (int t = 0; t < 4; ++t) {
      v2f b = ldB(Wqkv, NQKV, k0, col0 + 16 * t, lane);
      acc[t] = wmma4(a, b, acc[t]);
    }
  }
  const int g = lane >> 4;
#pragma unroll
  for (int t = 0; t < 4; ++t) {
    const int c = col0 + 16 * t + (lane & 15);
    const int three = c >> 9;          // 0=q 1=k 2=v
    const int head  = (c >> 6) & 7;
    const int dd    = c & 63;
    float* dst = (three == 0) ? Q : (three == 1) ? K : V;
    const float s = (three == 0) ? QSCALE : 1.0f;
#pragma unroll
    for (int v = 0; v < 8; ++v) {
      const int row  = row0 + v + 8 * g;      // 0..5119
      const int bidx = row / SEQ;
      const int nidx = row - bidx * SEQ;
      dst[(((size_t)bidx * HEADS + head) * SEQ + nidx) * DHEAD + dd] = acc[t][v] * s;
    }
  }
}

// =====================================================================
// Kernel 2: text causal attention. 1 wave per (16-row tile, bh).
// grid (16, 32), block 32. Scores in LDS, WMMA for QK^T and attn@V.
// =====================================================================
__global__ __launch_bounds__(32)
void k_text(const float* __restrict__ Q, const float* __restrict__ Kb,
            const float* __restrict__ Vb, const unsigned char* __restrict__ mask,
            float* __restrict__ ctx) {
  __shared__ float sc[16 * TEXTLEN];
  const int lane = threadIdx.x;
  const int row0 = blockIdx.x * 16;
  const int bh   = blockIdx.y;
  const float* q  = Q  + (size_t)bh * SEQ * DHEAD;
  const float* kp = Kb + (size_t)bh * SEQ * DHEAD;
  const float* vp = Vb + (size_t)bh * SEQ * DHEAD;

  // ---- scores: QK^T, one 16x16 tile at a time ----
  for (int ct = 0; ct < TEXTLEN / 16; ++ct) {
    v8f acc = {};
    const int col0 = ct * 16;
    for (int k0 = 0; k0 < DHEAD; k0 += 4) {
      v2f a = ldA(q, DHEAD, row0, k0, lane);
      v2f b = ldBT(kp, DHEAD, k0, col0, lane);   // B[k][j] = K[j][k]
      acc = wmma4(a, b, acc);
    }
    const int g = lane >> 4;
    const int c = col0 + (lane & 15);
#pragma unroll
    for (int v = 0; v < 8; ++v) sc[(v + 8 * g) * TEXTLEN + c] = acc[v];
  }
  __syncthreads();

  // ---- masked softmax per row ----
  const unsigned char* m = mask + (size_t)(bh >> 3) * TEXTLEN;
  for (int r = 0; r < 16; ++r) {
    const int grow = row0 + r;
    float mx = NEGMAX;
    for (int j = lane; j < TEXTLEN; j += 32) {
      float s = sc[r * TEXTLEN + j];
      if (j > grow || m[j] == 0) s = NEGMAX;
      sc[r * TEXTLEN + j] = s;
      mx = fmaxf(mx, s);
    }
    for (int off = 16; off; off >>= 1) mx = fmaxf(mx, __shfl_xor(mx, off, 32));
    float sum = 0.f;
    for (int j = lane; j < TEXTLEN; j += 32) {
      float e = expf(sc[r * TEXTLEN + j] - mx);
      sc[r * TEXTLEN + j] = e;
      sum += e;
    }
    for (int off = 16; off; off >>= 1) sum += __shfl_xor(sum, off, 32);
    const float inv = 1.f / sum;
    for (int j = lane; j < TEXTLEN; j += 32) sc[r * TEXTLEN + j] *= inv;
  }
  __syncthreads();

  // ---- out = attn @ V_text, 16x64 via 4 WMMA col tiles ----
  const int h = bh & 7, bidx = bh >> 3;
  const int g = lane >> 4;
  for (int t = 0; t < 4; ++t) {
    v8f acc = {};
    const int col0 = t * 16;
    for (int k0 = 0; k0 < TEXTLEN; k0 += 4) {
      v2f a;  // A from LDS scores (keep addrspace(3) by indexing sc directly)
      {
        const int rr = lane & 15, kk0 = k0 + 2 * g;
        a.x = sc[rr * TEXTLEN + kk0];
        a.y = sc[rr * TEXTLEN + kk0 + 1];
      }
      v2f b = ldB(vp, DHEAD, k0, col0, lane);
      acc = wmma4(a, b, acc);
    }
    const int d = col0 + (lane & 15);
#pragma unroll
    for (int v = 0; v < 8; ++v) {
      const int nrow = row0 + v + 8 * g;   // text row
      ctx[(((size_t)bidx * SEQ + nrow) * HEADS + h) * DHEAD + d] = acc[v];
    }
  }
}

// =====================================================================
// Kernel 3: image attention (full-to-text + 5x5 causal local).
// grid (64, 32), block 32 (1 wave).  Score row = 256 text + 25 local.
// =====================================================================
#define LROW 288   // padded score row stride (281 used)
__global__ __launch_bounds__(32)
void k_img(const float* __restrict__ Q, const float* __restrict__ Kb,
           const float* __restrict__ Vb, const unsigned char* __restrict__ mask,
           float* __restrict__ ctx) {
  __shared__ float sc[16 * LROW];
  __shared__ int   kidx[16 * KK];
  const int lane = threadIdx.x;
  const int row0 = blockIdx.x * 16;       // image-query tile base (0..1008)
  const int bh   = blockIdx.y;
  const float* q  = Q  + (size_t)bh * SEQ * DHEAD;
  const float* kp = Kb + (size_t)bh * SEQ * DHEAD;
  const float* vp = Vb + (size_t)bh * SEQ * DHEAD;

  // ---- image->text scores via WMMA ----
  for (int ct = 0; ct < TEXTLEN / 16; ++ct) {
    v8f acc = {};
    const int col0 = ct * 16;
    for (int k0 = 0; k0 < DHEAD; k0 += 4) {
      v2f a = ldA(q, DHEAD, TEXTLEN + row0, k0, lane);
      v2f b = ldBT(kp, DHEAD, k0, col0, lane);
      acc = wmma4(a, b, acc);
    }
    const int g = lane >> 4;
    const int c = col0 + (lane & 15);
#pragma unroll
    for (int v = 0; v < 8; ++v) sc[(v + 8 * g) * LROW + c] = acc[v];
  }

  // ---- local 5x5 scores (VALU dot products), causal + OOB masked ----
  for (int idx = lane; idx < 16 * KK; idx += 32) {
    const int r  = idx / KK, jj = idx - r * KK;
    const int p  = row0 + r;
    const int py = p >> 5, px = p & 31;
    const int ky = py + (jj / 5) - PADR;
    const int kx = px + (jj % 5) - PADR;
    float val = NEGMAX;
    int   kpos = 0;
    if (ky >= 0 && ky < IMGSZ && kx >= 0 && kx < IMGSZ) {
      kpos = ky * IMGSZ + kx;
      if (kpos <= p) {   // causal: allowed iff k_idx <= q_idx
        const float* qr = q  + (size_t)(TEXTLEN + p)    * DHEAD;
        const float* kr = kp + (size_t)(TEXTLEN + kpos) * DHEAD;
        float s = 0.f;
#pragma unroll 8
        for (int d = 0; d < DHEAD; ++d) s += qr[d] * kr[d];
        val = s;
      }
    }
    sc[r * LROW + TEXTLEN + jj] = val;   // masked taps get NEGMAX -> weight 0
    kidx[r * KK + jj] = kpos;            // clamped; weight 0 makes it harmless
  }
  __syncthreads();

  // ---- softmax over 281 entries per row (pad mask on text part) ----
  const unsigned char* m = mask + (size_t)(bh >> 3) * TEXTLEN;
  for (int r = 0; r < 16; ++r) {
    float mx = NEGMAX;
    for (int j = lane; j < TEXTLEN + KK; j += 32) {
      float s = sc[r * LROW + j];
      if (j < TEXTLEN && m[j] == 0) s = NEGMAX;
      sc[r * LROW + j] = s;
      mx = fmaxf(mx, s);
    }
    for (int off = 16; off; off >>= 1) mx = fmaxf(mx, __shfl_xor(mx, off, 32));
    float sum = 0.f;
    for (int j = lane; j < TEXTLEN + KK; j += 32) {
      float e = expf(sc[r * LROW + j] - mx);
      sc[r * LROW + j] = e;
      sum += e;
    }
    for (int off = 16; off; off >>= 1) sum += __shfl_xor(sum, off, 32);
    const float inv = 1.f / sum;
    for (int j = lane; j < TEXTLEN + KK; j += 32) sc[r * LROW + j] *= inv;
  }
  __syncthreads();

  // ---- out = attn_i2t @ V_text (WMMA) + local gather (VALU) ----
  const int h = bh & 7, bidx = bh >> 3;
  const int g = lane >> 4;
  for (int t = 0; t < 4; ++t) {
    v8f acc = {};
    const int col0 = t * 16;
    for (int k0 = 0; k0 < TEXTLEN; k0 += 4) {
      v2f a;
      {
        const int rr = lane & 15, kk0 = k0 + 2 * g;
        a.x = sc[rr * LROW + kk0];
        a.y = sc[rr * LROW + kk0 + 1];
      }
      v2f b = ldB(vp, DHEAD, k0, col0, lane);
      acc = wmma4(a, b, acc);
    }
    const int d = col0 + (lane & 15);
#pragma unroll
    for (int v = 0; v < 8; ++v) {
      const int r = v + 8 * g;
      const int p = row0 + r;
      float s = acc[v];
#pragma unroll
      for (int jj = 0; jj < KK; ++jj) {
        const float aw = sc[r * LROW + TEXTLEN + jj];
        const int  kpos = kidx[r * KK + jj];
        s += aw * vp[(size_t)(TEXTLEN + kpos) * DHEAD + d];
      }
      ctx[(((size_t)bidx * SEQ + (TEXTLEN + p)) * HEADS + h) * DHEAD + d] = s;
    }
  }
}

// =====================================================================
// Kernel 4: output projection. ctx[5120,512] @ Wout[512,512] + b_out.
// grid (40, 8), block 256.
// =====================================================================
__global__ __launch_bounds__(256)
void k_proj(const float* __restrict__ ctx, const float* __restrict__ Wout,
            const float* __restrict__ bout, float* __restrict__ out) {
  const int lane = threadIdx.x & 31;
  const int wave = threadIdx.x >> 5;
  const int row0 = blockIdx.x * 128 + wave * 16;
  const int col0 = blockIdx.y * 64;

  v8f acc[4] = {v8f{}, v8f{}, v8f{}, v8f{}};
  for (int k0 = 0; k0 < INNER; k0 += 4) {
    v2f a = ldA(ctx, INNER, row0, k0, lane);
#pragma unroll
    for (int t = 0; t < 4; ++t) {
      v2f b = ldB(Wout, DIM, k0, col0 + 16 * t, lane);
      acc[t] = wmma4(a, b, acc[t]);
    }
  }
  const int g = lane >> 4;
#pragma unroll
  for (int t = 0; t < 4; ++t) {
    const int c = col0 + 16 * t + (lane & 15);
    const float bv = bout[c];
#pragma unroll
    for (int v = 0; v < 8; ++v) {
      const int row = row0 + v + 8 * g;
      out[(size_t)row * DIM + c] = acc[t][v] + bv;
    }
  }
}

// =====================================================================
extern "C" void kernel_launch(void* const* d_in, const int* in_sizes, int n_in,
                              void* d_out, int out_size, void* d_ws, size_t ws_size,
                              hipStream_t stream) {
  (void)in_sizes; (void)n_in; (void)out_size; (void)ws_size;
  const float*         x    = (const float*)d_in[0];
  const unsigned char* mask = (const unsigned char*)d_in[1];   // jnp bool -> 1 byte
  const float*         Wq   = (const float*)d_in[2];
  const float*         Wo   = (const float*)d_in[3];
  const float*         bo   = (const float*)d_in[4];
  float*               out  = (float*)d_out;

  const size_t per = (size_t)BH * SEQ * DHEAD;   // 2,621,440 floats
  float* Q   = (float*)d_ws;
  float* K   = Q + per;
  float* V   = K + per;
  float* CTX = V + per;                          // [b][n][h*d], 2,621,440 floats

  k_qkv <<<dim3(40, 24), 256, 0, stream>>>(x, Wq, Q, K, V);
  k_text<<<dim3(TEXTLEN / 16, BH), 32, 0, stream>>>(Q, K, V, mask, CTX);
  k_img <<<dim3(IMGLEN / 16, BH),  32, 0, stream>>>(Q, K, V, mask, CTX);
  k_proj<<<dim3(40, 8), 256, 0, stream>>>(CTX, Wo, bo, out);
}